// GaussianSpot_10797547782660
// MI455X (gfx1250) — compile-verified
//
#include <hip/hip_runtime.h>
#include <hip/hip_bf16.h>

typedef __attribute__((ext_vector_type(2))) float v2f;
typedef __attribute__((ext_vector_type(8))) float v8f;

#define BB 256
#define FF 512
#define KK 2
#define DD 14

__global__ __launch_bounds__(256) void GaussianSpot_wmma_kernel(
    const int* __restrict__ batch_idx,
    const unsigned char* __restrict__ m_mask,     // bool, 1 byte/elem
    const float* __restrict__ height,
    const float* __restrict__ width,
    const float* __restrict__ x0,
    const float* __restrict__ y0,
    const float* __restrict__ background,
    const float* __restrict__ target_locs,        // (N, F, 1,1,1, 2)
    float* __restrict__ out)                      // (B, F, D, D, 1)
{
    const int lane = threadIdx.x & 31;
    const int wv   = threadIdx.x >> 5;
    // one wave per (b,f) tile; force wave-uniform so param loads scalarize
    const int tile = __builtin_amdgcn_readfirstlane((int)blockIdx.x * 8 + wv);
    const int b = tile >> 9;         // / FF (FF = 512)
    const int f = tile & (FF - 1);   // % FF

    const int pbase = tile * KK;

    // ---- wave-uniform parameter loads (K=2 spots) ----
    const float h0 = m_mask[pbase + 0] ? height[pbase + 0] : 0.0f;
    const float h1 = m_mask[pbase + 1] ? height[pbase + 1] : 0.0f;
    const float w0 = width[pbase + 0];
    const float w1 = width[pbase + 1];
    const float var0 = w0 * w0;
    const float var1 = w1 * w1;

    const int bi = batch_idx[b];
    const float* tl = target_locs + ((size_t)bi * FF + f) * 2;
    const float tlx = tl[0];
    const float tly = tl[1];
    const float lx0 = tlx + x0[pbase + 0], lx1 = tlx + x0[pbase + 1];
    const float ly0 = tly + y0[pbase + 0], ly1 = tly + y0[pbase + 1];

    const float bg = background[tile];

    // ---- per-lane separable Gaussian factors ----
    const int   r  = lane & 15;       // M index for A, N index for B
    const bool  lo = (lane < 16);     // lanes 16..31 carry K=2,3 -> zero
    const float p  = (float)r;

    const float TWO_PI = 6.28318530717958647692f;
    const float ninv0 = -0.5f / var0;
    const float ninv1 = -0.5f / var1;
    const float amp0  = h0 / (TWO_PI * var0);   // fold h and 1/(2*pi*var) into A
    const float amp1  = h1 / (TWO_PI * var1);

    const float dx0 = p - lx0, dx1 = p - lx1;
    const float dy0 = p - ly0, dy1 = p - ly1;

    const float gx0 = amp0 * __expf(dx0 * dx0 * ninv0);
    const float gx1 = amp1 * __expf(dx1 * dx1 * ninv1);
    const float gy0 = __expf(dy0 * dy0 * ninv0);
    const float gy1 = __expf(dy1 * dy1 * ninv1);

    // A: 16x4 f32 (lanes 0-15: VGPR0=K0, VGPR1=K1; lanes 16-31: K2,K3 = 0)
    v2f a, bm;
    a[0]  = lo ? gx0 : 0.0f;
    a[1]  = lo ? gx1 : 0.0f;
    // B: 4x16 f32, dual layout; rows K=2,3 zeroed
    bm[0] = lo ? gy0 : 0.0f;
    bm[1] = lo ? gy1 : 0.0f;

    // C preloaded with background: D = A x B + bg
    v8f c;
#pragma unroll
    for (int t = 0; t < 8; ++t) c[t] = bg;

    // EXEC is all-1s here (no divergence above; selects are v_cndmask)
    c = __builtin_amdgcn_wmma_f32_16x16x4_f32(
            /*neg_a=*/false, a, /*neg_b=*/false, bm,
            /*c_mod=*/(short)0, c, /*reuse_a=*/false, /*reuse_b=*/false);

    // ---- store valid 14x14 window of the 16x16 D tile ----
    // D layout: VGPR t, lanes 0-15 -> M=t, N=lane; lanes 16-31 -> M=t+8, N=lane-16
    float* __restrict__ outp = out + (size_t)tile * (DD * DD);
    const int j  = r;                 // py index
    const int ib = (lane >> 4) * 8;   // px base
    if (j < DD) {
#pragma unroll
        for (int t = 0; t < 8; ++t) {
            const int i = ib + t;
            if (i < DD) outp[i * DD + j] = c[t];
        }
    }
}

extern "C" void kernel_launch(void* const* d_in, const int* in_sizes, int n_in,
                              void* d_out, int out_size, void* d_ws, size_t ws_size,
                              hipStream_t stream) {
    const int*           batch_idx   = (const int*)d_in[0];
    const unsigned char* m_mask      = (const unsigned char*)d_in[1];
    const float*         height      = (const float*)d_in[2];
    const float*         width       = (const float*)d_in[3];
    const float*         x0          = (const float*)d_in[4];
    const float*         y0          = (const float*)d_in[5];
    const float*         background  = (const float*)d_in[6];
    const float*         target_locs = (const float*)d_in[7];
    float* out = (float*)d_out;

    const int tiles  = BB * FF;           // 131072 waves, one per (b,f)
    const int blocks = tiles / 8;         // 256 threads = 8 waves per block
    GaussianSpot_wmma_kernel<<<blocks, 256, 0, stream>>>(
        batch_idx, m_mask, height, width, x0, y0, background, target_locs, out);
}